// GSL_39908836114806
// MI455X (gfx1250) — compile-verified
//
#include <hip/hip_runtime.h>
#include <math.h>

#define N_NODES 8192
#define DIM     128
#define NBLK    64          // 8192 / 128 tile blocks per dim
#define NPART   (NBLK*NBLK) // 4096 per-block partials

typedef __attribute__((ext_vector_type(2))) float v2f;
typedef __attribute__((ext_vector_type(8))) float v8f;

__device__ __forceinline__ v8f wmma_f32(v2f a, v2f b, v8f c) {
    // V_WMMA_F32_16X16X4_F32 : D = A(16x4) * B(4x16) + C(16x16)
    return __builtin_amdgcn_wmma_f32_16x16x4_f32(
        /*neg_a=*/false, a, /*neg_b=*/false, b,
        /*c_mod=*/(short)0, c, /*reuse_a=*/false, /*reuse_b=*/false);
}

// ---------------------------------------------------------------------------
// Kernel 1: Y = feature (8192x128) @ W (128x128), f32 WMMA 16x16x4.
// Block = 256 threads (8 waves), each wave computes one 16x16 tile of Y.
// grid = (8, 64): blockIdx.x = N tile (0..7), blockIdx.y covers 8 M-tiles.
// ---------------------------------------------------------------------------
__global__ void gemmY(const float* __restrict__ F, const float* __restrict__ W,
                      float* __restrict__ Y) {
    const int lane = threadIdx.x & 31;
    const int wave = threadIdx.x >> 5;
    const int li = lane & 15, lh = lane >> 4;
    const int mTile = blockIdx.y * 8 + wave;   // 0..511
    const int nTile = blockIdx.x;              // 0..7
    const int col = nTile * 16 + li;

    const float* Arow = F + (size_t)(mTile * 16 + li) * DIM;
    v8f acc = {};
    #pragma unroll 4
    for (int k = 0; k < DIM; k += 4) {
        const int kk = k + lh * 2;
        v2f a = *(const v2f*)(Arow + kk);           // A: M=li, K=kk,kk+1
        v2f b;
        b.x = W[(size_t)kk * DIM + col];            // B: K=kk,  N=li
        b.y = W[(size_t)(kk + 1) * DIM + col];      // B: K=kk+1,N=li
        acc = wmma_f32(a, b, acc);
    }
    #pragma unroll
    for (int v = 0; v < 8; ++v) {
        const int i = mTile * 16 + v + lh * 8;      // C/D layout: M = v + 8*hi
        Y[(size_t)i * DIM + col] = acc[v];
    }
}

// ---------------------------------------------------------------------------
// Kernel 2: sq[i] = sum_k Y[i][k]^2.  One wave per row, one float4 per lane.
// ---------------------------------------------------------------------------
__global__ void rowSq(const float* __restrict__ Y, float* __restrict__ sq) {
    const int lane = threadIdx.x & 31;
    const int wave = threadIdx.x >> 5;
    const int row  = blockIdx.x * 8 + wave;
    float4 q = ((const float4*)(Y + (size_t)row * DIM))[lane];
    float s = q.x * q.x + q.y * q.y + q.z * q.z + q.w * q.w;
    #pragma unroll
    for (int o = 16; o > 0; o >>= 1) s += __shfl_down(s, o, 32);
    if (lane == 0) sq[row] = s;
}

// ---------------------------------------------------------------------------
// Kernel 3 (templated): recompute G-tile = Y Y^T via WMMA, then:
//   MODE 0: x = sqrt(relu(sq_i+sq_j-2g)); per-block partial (sum x, sum x^2)
//   MODE 1: e = exp(-x * c0);            per-block partial (sum e^2)
//   MODE 2: out[i,j] = e * inv           (the ONLY pass that touches HBM out)
// 128x128 block tile, 8 waves, each wave = 2x4 WMMA tiles (32x64 elements).
// ---------------------------------------------------------------------------
template <int MODE>
__global__ void pairK(const float* __restrict__ Y, const float* __restrict__ sq,
                      float* __restrict__ out, double* __restrict__ pA,
                      double* __restrict__ pB, const float* __restrict__ sc) {
    const int lane = threadIdx.x & 31;
    const int wave = threadIdx.x >> 5;
    const int li = lane & 15, lh = lane >> 4;
    const int jB = blockIdx.x, iB = blockIdx.y;
    const int wr = wave & 3, wc = wave >> 2;
    const int i0 = iB * 128 + wr * 32;   // rows  i0..i0+31 (2 tiles)
    const int j0 = jB * 128 + wc * 64;   // cols  j0..j0+63 (4 tiles)

    const float* Ai[2] = { Y + (size_t)(i0      + li) * DIM,
                           Y + (size_t)(i0 + 16 + li) * DIM };
    const float* Bj[4] = { Y + (size_t)(j0      + li) * DIM,
                           Y + (size_t)(j0 + 16 + li) * DIM,
                           Y + (size_t)(j0 + 32 + li) * DIM,
                           Y + (size_t)(j0 + 48 + li) * DIM };

    v8f acc[2][4];
    #pragma unroll
    for (int r = 0; r < 2; ++r)
        #pragma unroll
        for (int c = 0; c < 4; ++c) acc[r][c] = (v8f){};

    #pragma unroll 2
    for (int k = 0; k < DIM; k += 4) {
        const int kk = k + lh * 2;
        v2f a0 = *(const v2f*)(Ai[0] + kk);
        v2f a1 = *(const v2f*)(Ai[1] + kk);
        v2f b0 = *(const v2f*)(Bj[0] + kk);
        v2f b1 = *(const v2f*)(Bj[1] + kk);
        v2f b2 = *(const v2f*)(Bj[2] + kk);
        v2f b3 = *(const v2f*)(Bj[3] + kk);
        acc[0][0] = wmma_f32(a0, b0, acc[0][0]);
        acc[0][1] = wmma_f32(a0, b1, acc[0][1]);
        acc[0][2] = wmma_f32(a0, b2, acc[0][2]);
        acc[0][3] = wmma_f32(a0, b3, acc[0][3]);
        acc[1][0] = wmma_f32(a1, b0, acc[1][0]);
        acc[1][1] = wmma_f32(a1, b1, acc[1][1]);
        acc[1][2] = wmma_f32(a1, b2, acc[1][2]);
        acc[1][3] = wmma_f32(a1, b3, acc[1][3]);
    }

    float c0 = 0.0f, inv = 1.0f;
    if (MODE >= 1) c0 = sc[0];
    if (MODE == 2) inv = sc[1];

    // Hoist the sq loads (L2/L1-resident, 32KB table).
    float sqi[2][8], sqj[4];
    #pragma unroll
    for (int r = 0; r < 2; ++r)
        #pragma unroll
        for (int v = 0; v < 8; ++v)
            sqi[r][v] = sq[i0 + r * 16 + v + lh * 8];
    #pragma unroll
    for (int c = 0; c < 4; ++c) sqj[c] = sq[j0 + c * 16 + li];

    float lsum = 0.0f, lss = 0.0f;
    #pragma unroll
    for (int r = 0; r < 2; ++r) {
        #pragma unroll
        for (int c = 0; c < 4; ++c) {
            #pragma unroll
            for (int v = 0; v < 8; ++v) {
                const float g = acc[r][c][v];
                const float s = sqi[r][v] + sqj[c] - 2.0f * g;
                const float x = sqrtf(fmaxf(s, 0.0f));
                if (MODE == 0) {
                    lsum += x; lss += x * x;
                } else {
                    const float e = expf(-x * c0);
                    if (MODE == 1) {
                        lsum += e * e;
                    } else {
                        const int i = i0 + r * 16 + v + lh * 8;
                        const int j = j0 + c * 16 + li;
                        out[(size_t)i * N_NODES + j] = e * inv;
                    }
                }
            }
        }
    }

    if (MODE <= 1) {
        // Fixed-order lane reduce (deterministic), then block reduce via LDS.
        #pragma unroll
        for (int o = 16; o > 0; o >>= 1) {
            lsum += __shfl_down(lsum, o, 32);
            if (MODE == 0) lss += __shfl_down(lss, o, 32);
        }
        __shared__ float sA[8], sB[8];
        if (lane == 0) { sA[wave] = lsum; if (MODE == 0) sB[wave] = lss; }
        __syncthreads();
        if (threadIdx.x == 0) {
            double ta = 0.0, tb = 0.0;
            #pragma unroll
            for (int w = 0; w < 8; ++w) {
                ta += (double)sA[w];
                if (MODE == 0) tb += (double)sB[w];
            }
            const int blin = iB * gridDim.x + jB;
            pA[blin] = ta;
            if (MODE == 0) pB[blin] = tb;
        }
    }
}

// ---------------------------------------------------------------------------
// Fixed-order reductions over the 4096 per-block partials.
// ---------------------------------------------------------------------------
__global__ void statsSigma(const double* __restrict__ pA,
                           const double* __restrict__ pB,
                           float* __restrict__ sc) {
    __shared__ double sa[256], sb[256];
    const int t = threadIdx.x;
    double a = 0.0, b = 0.0;
    for (int i = 0; i < NPART / 256; ++i) {
        a += pA[t * (NPART / 256) + i];
        b += pB[t * (NPART / 256) + i];
    }
    sa[t] = a; sb[t] = b; __syncthreads();
    for (int o = 128; o > 0; o >>= 1) {
        if (t < o) { sa[t] += sa[t + o]; sb[t] += sb[t + o]; }
        __syncthreads();
    }
    if (t == 0) {
        const double n = (double)N_NODES * (double)N_NODES;
        const double var = (sb[0] - sa[0] * sa[0] / n) / (n - 1.0); // ddof=1
        sc[0] = (float)(0.5 * var);   // exp(-x * 0.5 * sigma^2)
    }
}

__global__ void statsNorm(const double* __restrict__ pC, float* __restrict__ sc) {
    __shared__ double sa[256];
    const int t = threadIdx.x;
    double a = 0.0;
    for (int i = 0; i < NPART / 256; ++i) a += pC[t * (NPART / 256) + i];
    sa[t] = a; __syncthreads();
    for (int o = 128; o > 0; o >>= 1) {
        if (t < o) sa[t] += sa[t + o];
        __syncthreads();
    }
    if (t == 0) {
        const double nrm = sqrt(sa[0]);
        sc[1] = (float)(1.0 / fmax(nrm, 1e-12));
    }
}

// ---------------------------------------------------------------------------
extern "C" void kernel_launch(void* const* d_in, const int* in_sizes, int n_in,
                              void* d_out, int out_size, void* d_ws, size_t ws_size,
                              hipStream_t stream) {
    const float* feature = (const float*)d_in[0];   // [8192,128]
    const float* weight  = (const float*)d_in[1];   // [128,128]
    float* out = (float*)d_out;                     // [8192,8192]

    char* ws = (char*)d_ws;
    float*  Y   = (float*)(ws);                                   // 4 MB
    float*  sq  = (float*)(ws + (size_t)4194304);                 // 32 KB
    double* pA  = (double*)(ws + (size_t)4227072);                // 32 KB
    double* pB  = (double*)(ws + (size_t)4259840);                // 32 KB
    double* pC  = (double*)(ws + (size_t)4292608);                // 32 KB
    float*  sc  = (float*)(ws + (size_t)4325376);                 // scalars

    gemmY<<<dim3(8, 64), 256, 0, stream>>>(feature, weight, Y);
    rowSq<<<dim3(1024), 256, 0, stream>>>(Y, sq);

    // Pass 1: stats for sigma (no HBM store of the 268MB matrix).
    pairK<0><<<dim3(NBLK, NBLK), 256, 0, stream>>>(Y, sq, nullptr, pA, pB, sc);
    statsSigma<<<1, 256, 0, stream>>>(pA, pB, sc);

    // Pass 2: recompute, accumulate sum(exp^2) for the global L2 norm.
    pairK<1><<<dim3(NBLK, NBLK), 256, 0, stream>>>(Y, sq, nullptr, pC, pB, sc);
    statsNorm<<<1, 256, 0, stream>>>(pC, sc);

    // Pass 3: recompute, write final normalized output (single 268MB write).
    pairK<2><<<dim3(NBLK, NBLK), 256, 0, stream>>>(Y, sq, out, pA, pB, sc);
}